// ConditionalProbabilityEstimator_25056839205461
// MI455X (gfx1250) — compile-verified
//
#include <hip/hip_runtime.h>
#include <hip/hip_bf16.h>
#include <stdint.h>

// Count-Min Sketch: D=2 hash rows, W=2^26 buckets, multiply-shift hashing.
namespace {
constexpr uint32_t kWLog  = 26u;
constexpr uint32_t kW     = 1u << kWLog;        // 67,108,864 buckets per row
constexpr uint32_t kShift = 32u - kWLog;        // keep top 26 bits
constexpr int      kVec   = 4;                  // elements per thread (main kernels)

__device__ __forceinline__ uint32_t cms_hash(uint32_t x, uint32_t a, uint32_t b) {
    // multiply-shift universal hash; uint32 wrap == jnp uint32 semantics
    return (a * x + b) >> kShift;
}
} // namespace

// ---------------------------------------------------------------------------
// UPDATE (main): exactly 4 ids/thread, no bounds checks. Two global_load_b128
// for the id streams, 16 hash VALUs, 16 global_atomic_add_f32 (no-return ->
// STOREcnt) random scatters. inc=+1 builds the sketch, inc=-1 is the exact
// undo used only by the restore fallback path.
// ---------------------------------------------------------------------------
__global__ __launch_bounds__(256) void cms_update_main(
    const int* __restrict__ qid, const int* __restrict__ pid,
    float* __restrict__ dqp, float* __restrict__ dq,
    const uint32_t* __restrict__ ha, const uint32_t* __restrict__ hb,
    float inc)
{
    const uint32_t a0 = ha[0], a1 = ha[1];      // uniform -> s_load
    const uint32_t b0 = hb[0], b1 = hb[1];

    const int base = (blockIdx.x * blockDim.x + threadIdx.x) * kVec;
    const int4 qv = *reinterpret_cast<const int4*>(qid + base);   // b128
    const int4 pv = *reinterpret_cast<const int4*>(pid + base);   // b128

    const uint32_t q[kVec] = {(uint32_t)qv.x, (uint32_t)qv.y, (uint32_t)qv.z, (uint32_t)qv.w};
    const uint32_t p[kVec] = {(uint32_t)pv.x, (uint32_t)pv.y, (uint32_t)pv.z, (uint32_t)pv.w};

    #pragma unroll
    for (int k = 0; k < kVec; ++k) {
        const uint32_t qp = p[k] + 17u * q[k];          // int32 wrap == uint32 wrap
        atomicAdd(&dqp[cms_hash(qp,   a0, b0)],      inc);
        atomicAdd(&dqp[kW + cms_hash(qp, a1, b1)],   inc);
        atomicAdd(&dq [cms_hash(q[k], a0, b0)],      inc);
        atomicAdd(&dq [kW + cms_hash(q[k], a1, b1)], inc);
    }
}

// UPDATE (tail): one id per thread, bounds-checked; launched only if n % 1024 != 0.
__global__ __launch_bounds__(64) void cms_update_tail(
    const int* __restrict__ qid, const int* __restrict__ pid,
    float* __restrict__ dqp, float* __restrict__ dq,
    const uint32_t* __restrict__ ha, const uint32_t* __restrict__ hb,
    int start, int n, float inc)
{
    const int i = start + (int)threadIdx.x;
    if (i >= n) return;
    const uint32_t a0 = ha[0], a1 = ha[1], b0 = hb[0], b1 = hb[1];
    const uint32_t q = (uint32_t)qid[i];
    const uint32_t p = (uint32_t)pid[i];
    const uint32_t qp = p + 17u * q;
    atomicAdd(&dqp[cms_hash(qp, a0, b0)],      inc);
    atomicAdd(&dqp[kW + cms_hash(qp, a1, b1)], inc);
    atomicAdd(&dq [cms_hash(q,  a0, b0)],      inc);
    atomicAdd(&dq [kW + cms_hash(q,  a1, b1)], inc);
}

// ---------------------------------------------------------------------------
// Shared per-id query body: gather delta (+ base when separate), min across
// the two hash rows, emit all six reference outputs as float32 (ids < 2^24 so
// the int->float conversion is exact). Streaming NT stores for the output.
// ---------------------------------------------------------------------------
__device__ __forceinline__ void query_one(
    int i, uint32_t q, uint32_t p,
    const float* __restrict__ base_qp, const float* __restrict__ base_q,
    const float* __restrict__ dqp, const float* __restrict__ dq,
    uint32_t a0, uint32_t a1, uint32_t b0, uint32_t b1,
    float* __restrict__ out, int n, bool add_base)
{
    const uint32_t qp = p + 17u * q;

    const uint32_t hqp0 = cms_hash(qp, a0, b0);
    const uint32_t hqp1 = kW + cms_hash(qp, a1, b1);
    const uint32_t hq0  = cms_hash(q, a0, b0);
    const uint32_t hq1  = kW + cms_hash(q, a1, b1);

    float vqp0 = dqp[hqp0];
    float vqp1 = dqp[hqp1];
    float vq0  = dq[hq0];
    float vq1  = dq[hq1];
    if (add_base) {                       // delta tables live in workspace
        vqp0 += base_qp[hqp0];
        vqp1 += base_qp[hqp1];
        vq0  += base_q[hq0];
        vq1  += base_q[hq1];
    }
    const float qp_freq = fminf(vqp0, vqp1);
    const float q_freq  = fminf(vq0,  vq1);
    const float fq = (float)(int32_t)q;
    const float fp = (float)(int32_t)p;

    __builtin_nontemporal_store(qp_freq, out + i);                 // qp_freqs
    __builtin_nontemporal_store(q_freq,  out + (size_t)n + i);     // q_freqs
    __builtin_nontemporal_store(fq,      out + 2ull * n + i);      // q
    __builtin_nontemporal_store(fp,      out + 3ull * n + i);      // p
    __builtin_nontemporal_store(fq,      out + 4ull * n + i);      // all_q
    __builtin_nontemporal_store(fp,      out + 5ull * n + i);      // all_p
}

// QUERY (main): exactly 4 ids/thread, no bounds checks.
__global__ __launch_bounds__(256) void cms_query_main(
    const int* __restrict__ qid, const int* __restrict__ pid,
    const float* __restrict__ base_qp, const float* __restrict__ base_q,
    const float* __restrict__ dqp, const float* __restrict__ dq,
    const uint32_t* __restrict__ ha, const uint32_t* __restrict__ hb,
    float* __restrict__ out, int n, int add_base)
{
    const uint32_t a0 = ha[0], a1 = ha[1];
    const uint32_t b0 = hb[0], b1 = hb[1];

    const int base = (blockIdx.x * blockDim.x + threadIdx.x) * kVec;
    const int4 qv = *reinterpret_cast<const int4*>(qid + base);
    const int4 pv = *reinterpret_cast<const int4*>(pid + base);

    const uint32_t q[kVec] = {(uint32_t)qv.x, (uint32_t)qv.y, (uint32_t)qv.z, (uint32_t)qv.w};
    const uint32_t p[kVec] = {(uint32_t)pv.x, (uint32_t)pv.y, (uint32_t)pv.z, (uint32_t)pv.w};

    #pragma unroll
    for (int k = 0; k < kVec; ++k)
        query_one(base + k, q[k], p[k], base_qp, base_q, dqp, dq,
                  a0, a1, b0, b1, out, n, add_base != 0);
}

// QUERY (tail): one id per thread, bounds-checked; launched only if n % 1024 != 0.
__global__ __launch_bounds__(64) void cms_query_tail(
    const int* __restrict__ qid, const int* __restrict__ pid,
    const float* __restrict__ base_qp, const float* __restrict__ base_q,
    const float* __restrict__ dqp, const float* __restrict__ dq,
    const uint32_t* __restrict__ ha, const uint32_t* __restrict__ hb,
    float* __restrict__ out, int start, int n, int add_base)
{
    const int i = start + (int)threadIdx.x;
    if (i >= n) return;
    query_one(i, (uint32_t)qid[i], (uint32_t)pid[i], base_qp, base_q, dqp, dq,
              ha[0], ha[1], hb[0], hb[1], out, n, add_base != 0);
}

extern "C" void kernel_launch(void* const* d_in, const int* in_sizes, int n_in,
                              void* d_out, int out_size, void* d_ws, size_t ws_size,
                              hipStream_t stream) {
    // setup_inputs order: query_ids, pos_ids, sync, qp_table, q_table, ha, hb
    const int*      qid      = (const int*)d_in[0];
    const int*      pid      = (const int*)d_in[1];
    const float*    qp_table = (const float*)d_in[3];
    const float*    q_table  = (const float*)d_in[4];
    const uint32_t* ha       = (const uint32_t*)d_in[5];
    const uint32_t* hb       = (const uint32_t*)d_in[6];
    float*          out      = (float*)d_out;
    const int n = in_sizes[0];

    const size_t tbl_elems = 2ull * kW;                        // D * W per table
    const size_t need      = 2ull * tbl_elems * sizeof(float); // both delta tables: 1 GiB

    const int block   = 256;
    const int n_main  = (n / (block * kVec)) * (block * kVec); // full 1024-element blocks
    const int grid    = n_main / (block * kVec);
    const int tail    = n - n_main;                            // 128 for N = 2,000,000

    const bool use_ws = (ws_size >= need);
    float* dqp = use_ws ? (float*)d_ws             : (float*)qp_table;
    float* dq  = use_ws ? (float*)d_ws + tbl_elems : (float*)q_table;
    const int add_base = use_ws ? 1 : 0;

    if (use_ws)
        hipMemsetAsync(d_ws, 0, need, stream);     // graph-capturable memset node

    // Build sketch deltas.
    if (grid > 0)
        cms_update_main<<<grid, block, 0, stream>>>(qid, pid, dqp, dq, ha, hb, 1.0f);
    if (tail > 0)
        cms_update_tail<<<1, 64, 0, stream>>>(qid, pid, dqp, dq, ha, hb, n_main, n, 1.0f);

    // Query: min over rows of (base + delta).
    if (grid > 0)
        cms_query_main<<<grid, block, 0, stream>>>(qid, pid, qp_table, q_table,
                                                   dqp, dq, ha, hb, out, n, add_base);
    if (tail > 0)
        cms_query_tail<<<1, 64, 0, stream>>>(qid, pid, qp_table, q_table,
                                             dqp, dq, ha, hb, out, n_main, n, add_base);

    // Fallback only: exactly undo the +1.0 increments so the input tables are
    // restored bit-identically (small-integer float add/sub is exact).
    if (!use_ws) {
        if (grid > 0)
            cms_update_main<<<grid, block, 0, stream>>>(qid, pid, dqp, dq, ha, hb, -1.0f);
        if (tail > 0)
            cms_update_tail<<<1, 64, 0, stream>>>(qid, pid, dqp, dq, ha, hb, n_main, n, -1.0f);
    }
}